// RandomWalkNet_68221260530017
// MI455X (gfx1250) — compile-verified
//
#include <hip/hip_runtime.h>
#include <hip/hip_bf16.h>

// ---------------------------------------------------------------------------
// RandomWalkNet re-ranking pipeline for gfx1250 (MI455X), f32 WMMA throughout.
//
//   f   = x @ Wb                                  (4096x512)
//   FT  = gallery(f)^T                            (512x3072)  LDS transpose
//   P0  = exp(-g0 * d2(probe, gallery))           (1024x3072) d2 via WMMA
//   GG0 = exp(-g0 * d2(gallery, gallery))         (3072x3072) d2 via WMMA
//   M   = I - 0.99 * GG0 / off_rowsum             (3072x3072)
//   solve M X = P0^T via blocked unpivoted LU (M strictly diag dominant)
//   pg0 = 0.01 * X^T
// ---------------------------------------------------------------------------

typedef __attribute__((ext_vector_type(2))) float v2f;
typedef __attribute__((ext_vector_type(8))) float v8f;

#define NROWS 4096
#define DDIM  512
#define CDIM  512
#define GDIM  3072   // gallery count
#define PDIM  1024   // probe count
#define NB    64     // LU block size
#define ALPHA_C 0.99f

__device__ __forceinline__ v8f wmma4(v2f a, v2f b, v8f c) {
  // D = A(16x4,f32) x B(4x16,f32) + C(16x16,f32)
  return __builtin_amdgcn_wmma_f32_16x16x4_f32(
      false, a, false, b, (short)0, c, false, false);
}

// gallery index -> row of f
__device__ __forceinline__ int grow(int j) { return (j / 3) * 4 + (j % 3) + 1; }

// ------------------------------ f = x @ Wb ---------------------------------
__global__ void gemm_f_kernel(const float* __restrict__ X,
                              const float* __restrict__ W,
                              float* __restrict__ F) {
  int wave = blockIdx.x * (blockDim.x >> 5) + (threadIdx.x >> 5);
  const int tilesN = CDIM / 16;                 // 32
  int m0 = (wave / tilesN) * 16;
  int n0 = (wave % tilesN) * 16;
  int lane = threadIdx.x & 31, half = lane >> 4, l15 = lane & 15;

  v8f acc = {0, 0, 0, 0, 0, 0, 0, 0};
  for (int kk = 0; kk < DDIM; kk += 4) {
    v2f a;
    a.x = X[(m0 + l15) * DDIM + kk + 2 * half];
    a.y = X[(m0 + l15) * DDIM + kk + 2 * half + 1];
    v2f b;
    b.x = W[(kk + half) * CDIM + n0 + l15];
    b.y = W[(kk + 2 + half) * CDIM + n0 + l15];
    acc = wmma4(a, b, acc);
  }
  int rbase = m0 + (half ? 8 : 0);
#pragma unroll
  for (int v = 0; v < 8; v++) F[(rbase + v) * CDIM + n0 + l15] = acc[v];
}

// ------------------ FT[k][j] = f[grow(j)][k]  (LDS transpose) --------------
__global__ void transpose_g_kernel(const float* __restrict__ F,
                                   float* __restrict__ FT) {
  __shared__ float tile[32][33];
  int j0 = blockIdx.x * 32;                     // gallery base
  int k0 = blockIdx.y * 32;                     // feature base
  int tx = threadIdx.x & 31, ty = threadIdx.x >> 5;   // 32 x 8
  for (int yy = ty; yy < 32; yy += 8)
    tile[yy][tx] = F[grow(j0 + yy) * CDIM + k0 + tx]; // coalesced along k
  __syncthreads();
  for (int yy = ty; yy < 32; yy += 8)
    FT[(k0 + yy) * GDIM + j0 + tx] = tile[tx][yy];    // coalesced along j
}

// ------------------------------ row norms ----------------------------------
__global__ void row_norms_kernel(const float* __restrict__ F,
                                 float* __restrict__ norms) {
  int r = blockIdx.x, t = threadIdx.x;
  __shared__ float red[256];
  float s = 0.f;
  for (int c = t; c < CDIM; c += 256) { float v = F[r * CDIM + c]; s += v * v; }
  red[t] = s; __syncthreads();
  for (int o = 128; o > 0; o >>= 1) { if (t < o) red[t] += red[t + o]; __syncthreads(); }
  if (t == 0) norms[r] = red[0];
}

// --------------- gallery x gallery^T + embed epilogue ----------------------
__global__ void gemm_gg_kernel(const float* __restrict__ F,
                               const float* __restrict__ FT,
                               const float* __restrict__ norms,
                               const float* __restrict__ gamma,
                               float* __restrict__ gg_out) {
  int wave = blockIdx.x * 8 + (threadIdx.x >> 5);
  const int tilesN = GDIM / 16;                 // 192
  int m0 = (wave / tilesN) * 16;
  int n0 = (wave % tilesN) * 16;
  int lane = threadIdx.x & 31, half = lane >> 4, l15 = lane & 15;

  int rowg = grow(m0 + l15);
  int colj = n0 + l15;
  v8f acc = {0, 0, 0, 0, 0, 0, 0, 0};
  for (int kk = 0; kk < CDIM; kk += 4) {
    v2f a;
    a.x = F[rowg * CDIM + kk + 2 * half];
    a.y = F[rowg * CDIM + kk + 2 * half + 1];
    v2f b;                                      // coalesced via FT
    b.x = FT[(kk + half) * GDIM + colj];
    b.y = FT[(kk + 2 + half) * GDIM + colj];
    acc = wmma4(a, b, acc);
  }
  float ncol = norms[grow(colj)];
  float g0 = gamma[0], g1 = gamma[1];
  int rbase = m0 + (half ? 8 : 0);
#pragma unroll
  for (int v = 0; v < 8; v++) {
    int r = rbase + v;
    float nr = norms[grow(r)];
    float d2 = (nr + ncol - 2.f * acc[v]) * (1.f / (float)CDIM);
    int o = (r * GDIM + colj) * 2;
    gg_out[o]     = expf(-g0 * d2);
    gg_out[o + 1] = g1 * d2;
  }
}

// --------------- probe x gallery^T + embed epilogue ------------------------
// writes p_g[..,1] into out; writes p_g0^T into B (rhs for solver)
__global__ void gemm_pg_kernel(const float* __restrict__ F,
                               const float* __restrict__ FT,
                               const float* __restrict__ norms,
                               const float* __restrict__ gamma,
                               float* __restrict__ pg_out,
                               float* __restrict__ Brhs) {
  int wave = blockIdx.x * 8 + (threadIdx.x >> 5);
  const int tilesN = GDIM / 16;                 // 192
  int m0 = (wave / tilesN) * 16;
  int n0 = (wave % tilesN) * 16;
  int lane = threadIdx.x & 31, half = lane >> 4, l15 = lane & 15;

  int rowf = 4 * (m0 + l15);                    // probe p -> f row 4p
  int colj = n0 + l15;
  v8f acc = {0, 0, 0, 0, 0, 0, 0, 0};
  for (int kk = 0; kk < CDIM; kk += 4) {
    v2f a;
    a.x = F[rowf * CDIM + kk + 2 * half];
    a.y = F[rowf * CDIM + kk + 2 * half + 1];
    v2f b;
    b.x = FT[(kk + half) * GDIM + colj];
    b.y = FT[(kk + 2 + half) * GDIM + colj];
    acc = wmma4(a, b, acc);
  }
  float ncol = norms[grow(colj)];
  float g0 = gamma[0], g1 = gamma[1];
  int rbase = m0 + (half ? 8 : 0);
#pragma unroll
  for (int v = 0; v < 8; v++) {
    int p = rbase + v;
    float nr = norms[4 * p];
    float d2 = (nr + ncol - 2.f * acc[v]) * (1.f / (float)CDIM);
    pg_out[(p * GDIM + colj) * 2 + 1] = g1 * d2;
    Brhs[colj * PDIM + p] = expf(-g0 * d2);     // B = pg0_init^T
  }
}

// ----------------- off-diagonal row sums of gg0 ----------------------------
__global__ void rowsum_kernel(const float* __restrict__ gg,
                              float* __restrict__ rowsum) {
  int i = blockIdx.x, t = threadIdx.x;
  __shared__ float red[256];
  float s = 0.f;
  for (int j = t; j < GDIM; j += 256) s += gg[(i * GDIM + j) * 2];
  red[t] = s; __syncthreads();
  for (int o = 128; o > 0; o >>= 1) { if (t < o) red[t] += red[t + o]; __syncthreads(); }
  if (t == 0) rowsum[i] = red[0] - gg[(i * GDIM + i) * 2];
}

// ----------------- M = I - alpha * gg0 / rowsum ----------------------------
__global__ void build_M_kernel(const float* __restrict__ gg,
                               const float* __restrict__ rowsum,
                               float* __restrict__ M) {
  int idx = blockIdx.x * 256 + threadIdx.x;
  if (idx >= GDIM * GDIM) return;
  int i = idx / GDIM, j = idx - i * GDIM;
  float a = gg[idx * 2] / rowsum[i];
  M[idx] = ((i == j) ? 1.f : 0.f) - ALPHA_C * a;
}

// ----------------- LU: factor NBxNB diagonal block in LDS ------------------
__global__ void lu_diag_kernel(float* __restrict__ M, int kb) {
  __shared__ float blk[NB][NB + 1];
  int t = threadIdx.x;  // NB threads, thread t owns row t
  for (int j = 0; j < NB; j++) blk[t][j] = M[(kb + t) * GDIM + kb + j];
  __syncthreads();
  for (int k = 0; k < NB - 1; k++) {
    if (t > k) {
      float l = blk[t][k] / blk[k][k];
      blk[t][k] = l;
      for (int j = k + 1; j < NB; j++) blk[t][j] -= l * blk[k][j];
    }
    __syncthreads();
  }
  for (int j = 0; j < NB; j++) M[(kb + t) * GDIM + kb + j] = blk[t][j];
}

// ----- LU panels (fused, independent):  y==0: L21 = A21 U11^-1 (row fwd)
//                                        y==1: U12 = L11^-1 A12 (col fwd)
__global__ void lu_panels_kernel(float* __restrict__ M, int kb) {
  __shared__ float D[NB][NB + 1];
  __shared__ float w[NB][NB];
  int t = threadIdx.x;  // NB threads
  for (int j = 0; j < NB; j++) D[j][t] = M[(kb + j) * GDIM + kb + t];
  __syncthreads();
  if (blockIdx.y == 0) {
    int i = kb + NB + blockIdx.x * NB + t;      // one row per thread
    for (int j = 0; j < NB; j++) {
      float s = M[i * GDIM + kb + j];
      for (int u = 0; u < j; u++) s -= w[u][t] * D[u][j];
      w[j][t] = s / D[j][j];
    }
    for (int j = 0; j < NB; j++) M[i * GDIM + kb + j] = w[j][t];
  } else {
    int col = kb + NB + blockIdx.x * NB + t;    // one column per thread
    for (int r = 0; r < NB; r++) {
      float s = M[(kb + r) * GDIM + col];
      for (int u = 0; u < r; u++) s -= D[r][u] * w[u][t];
      w[r][t] = s;
      M[(kb + r) * GDIM + col] = s;
    }
  }
}

// ----------------- forward substitution on RHS block -----------------------
__global__ void fs_diag_kernel(const float* __restrict__ M,
                               float* __restrict__ B, int kb) {
  __shared__ float L[NB][NB + 1];
  __shared__ float ys[NB][NB];
  int t = threadIdx.x;
  for (int j = 0; j < NB; j++) L[j][t] = M[(kb + j) * GDIM + kb + t];
  __syncthreads();
  int p = blockIdx.x * NB + t;
  for (int r = 0; r < NB; r++) {
    float s = B[(kb + r) * PDIM + p];
    for (int u = 0; u < r; u++) s -= L[r][u] * ys[u][t];
    ys[r][t] = s;
    B[(kb + r) * PDIM + p] = s;
  }
}

// ----------------- backward substitution on RHS block ----------------------
__global__ void bs_diag_kernel(const float* __restrict__ M,
                               float* __restrict__ B, int kb) {
  __shared__ float U[NB][NB + 1];
  __shared__ float xs[NB][NB];
  int t = threadIdx.x;
  for (int j = 0; j < NB; j++) U[j][t] = M[(kb + j) * GDIM + kb + t];
  __syncthreads();
  int p = blockIdx.x * NB + t;
  for (int r = NB - 1; r >= 0; r--) {
    float s = B[(kb + r) * PDIM + p];
    for (int u = r + 1; u < NB; u++) s -= U[r][u] * xs[u][t];
    s /= U[r][r];
    xs[r][t] = s;
    B[(kb + r) * PDIM + p] = s;
  }
}

// ----------------- generic C -= A @ B via WMMA (16x16 tiles) ---------------
__global__ void wmma_update_kernel(float* __restrict__ C, int ldc,
                                   const float* __restrict__ A, int lda,
                                   const float* __restrict__ Bm, int ldb,
                                   int Md, int Nd, int Kd) {
  int wave = blockIdx.x * (blockDim.x >> 5) + (threadIdx.x >> 5);
  int tilesN = Nd >> 4;
  int tiles = (Md >> 4) * tilesN;
  if (wave >= tiles) return;                    // wave-uniform guard
  int m0 = (wave / tilesN) << 4;
  int n0 = (wave % tilesN) << 4;
  int lane = threadIdx.x & 31, half = lane >> 4, l15 = lane & 15;
  int rbase = m0 + (half ? 8 : 0);

  v8f acc;
#pragma unroll
  for (int v = 0; v < 8; v++) acc[v] = C[(rbase + v) * ldc + n0 + l15];
  for (int kk = 0; kk < Kd; kk += 4) {
    v2f a;                                      // negate A: C + (-A)B == C - AB
    a.x = -A[(m0 + l15) * lda + kk + 2 * half];
    a.y = -A[(m0 + l15) * lda + kk + 2 * half + 1];
    v2f b;
    b.x = Bm[(kk + half) * ldb + n0 + l15];
    b.y = Bm[(kk + 2 + half) * ldb + n0 + l15];
    acc = wmma4(a, b, acc);
  }
#pragma unroll
  for (int v = 0; v < 8; v++) C[(rbase + v) * ldc + n0 + l15] = acc[v];
}

// ----------------- final: pg0 = (1-alpha) * X^T ----------------------------
__global__ void write_pg0_kernel(const float* __restrict__ B,
                                 float* __restrict__ out) {
  int idx = blockIdx.x * 256 + threadIdx.x;
  if (idx >= PDIM * GDIM) return;
  int p = idx / GDIM, j = idx - p * GDIM;
  out[idx * 2] = (1.f - ALPHA_C) * B[j * PDIM + p];
}

// ===========================================================================
extern "C" void kernel_launch(void* const* d_in, const int* in_sizes, int n_in,
                              void* d_out, int out_size, void* d_ws, size_t ws_size,
                              hipStream_t stream) {
  (void)in_sizes; (void)n_in; (void)out_size; (void)ws_size;
  const float* x     = (const float*)d_in[0];
  const float* Wb    = (const float*)d_in[1];
  const float* gamma = (const float*)d_in[2];
  float* out = (float*)d_out;

  // workspace layout (floats)
  float* ws     = (float*)d_ws;
  float* F      = ws;                                   // 4096*512
  float* norms  = F + NROWS * DDIM;                     // 4096
  float* rowsum = norms + NROWS;                        // 3072
  float* M      = rowsum + GDIM;                        // 3072*3072
  float* B      = M + GDIM * GDIM;                      // 3072*1024
  float* FT     = B + GDIM * PDIM;                      // 512*3072

  float* gg_out = out + PDIM * GDIM * 2;                // g_g section of score
  float* pg_out = out;                                  // p_g section of score

  // 1) f = x @ Wb          (256*32 = 8192 tiles, 8 waves/block)
  gemm_f_kernel<<<(NROWS / 16) * (CDIM / 16) / 8, 256, 0, stream>>>(x, Wb, F);

  // 2) gallery^T and row norms (independent of each other)
  transpose_g_kernel<<<dim3(GDIM / 32, CDIM / 32), 256, 0, stream>>>(F, FT);
  row_norms_kernel<<<NROWS, 256, 0, stream>>>(F, norms);

  // 3) g_g embed (writes both channels of g_g section)
  gemm_gg_kernel<<<(GDIM / 16) * (GDIM / 16) / 8, 256, 0, stream>>>(F, FT, norms, gamma, gg_out);

  // 4) off-diagonal row sums of gg0
  rowsum_kernel<<<GDIM, 256, 0, stream>>>(gg_out, rowsum);

  // 5) M = I - alpha * gg0 / rowsum
  build_M_kernel<<<(GDIM * GDIM + 255) / 256, 256, 0, stream>>>(gg_out, rowsum, M);

  // 6) p_g embed: channel 1 to out, pg0_init^T to B
  gemm_pg_kernel<<<(PDIM / 16) * (GDIM / 16) / 8, 256, 0, stream>>>(F, FT, norms, gamma, pg_out, B);

  // 7) blocked LU of M (no pivoting; M strictly diagonally dominant)
  for (int kb = 0; kb < GDIM; kb += NB) {
    lu_diag_kernel<<<1, NB, 0, stream>>>(M, kb);
    int rem = GDIM - kb - NB;
    if (rem > 0) {
      lu_panels_kernel<<<dim3(rem / NB, 2), NB, 0, stream>>>(M, kb);
      int tiles = (rem / 16) * (rem / 16);
      wmma_update_kernel<<<(tiles + 7) / 8, 256, 0, stream>>>(
          M + (kb + NB) * GDIM + (kb + NB), GDIM,
          M + (kb + NB) * GDIM + kb,        GDIM,
          M + kb * GDIM + (kb + NB),        GDIM,
          rem, rem, NB);
    }
  }

  // 8) forward solve L Y = B (unit lower)
  for (int kb = 0; kb < GDIM; kb += NB) {
    fs_diag_kernel<<<PDIM / NB, NB, 0, stream>>>(M, B, kb);
    int rem = GDIM - kb - NB;
    if (rem > 0) {
      int tiles = (rem / 16) * (PDIM / 16);
      wmma_update_kernel<<<(tiles + 7) / 8, 256, 0, stream>>>(
          B + (kb + NB) * PDIM, PDIM,
          M + (kb + NB) * GDIM + kb, GDIM,
          B + kb * PDIM, PDIM,
          rem, PDIM, NB);
    }
  }

  // 9) backward solve U X = Y
  for (int kb = GDIM - NB; kb >= 0; kb -= NB) {
    bs_diag_kernel<<<PDIM / NB, NB, 0, stream>>>(M, B, kb);
    if (kb > 0) {
      int tiles = (kb / 16) * (PDIM / 16);
      wmma_update_kernel<<<(tiles + 7) / 8, 256, 0, stream>>>(
          B, PDIM,
          M + kb, GDIM,
          B + kb * PDIM, PDIM,
          kb, PDIM, NB);
    }
  }

  // 10) pg0 = (1-alpha) * X^T into p_g channel 0
  write_pg0_kernel<<<(PDIM * GDIM + 255) / 256, 256, 0, stream>>>(B, out);
}